// BiAttention_76794015252634
// MI455X (gfx1250) — compile-verified
//
#include <hip/hip_runtime.h>
#include <math.h>

// BiAttention (mode==1) for MI455X / gfx1250, wave32, f32 WMMA 16x16x4.
// B=16, T=2048, Wn=512, D=300. Full f32 precision (bf16 would corrupt softmax logits).
//
// Edge tiles (N=300 = 18*16+12): B-side loads use a CLAMPED row/col (stay
// in-bounds) with NO value mask — WMMA columns are independent, so garbage in
// B column n only corrupts D column n, which the guarded store never writes.

typedef __attribute__((ext_vector_type(2))) float v2f;
typedef __attribute__((ext_vector_type(8))) float v8f;

#define Bc   16
#define Tc   2048
#define WNc  512
#define Dc   300
#define SST  516   // LDS score-row stride (floats): 516%64==4 -> conflict-free transposed reads

// ---------------------------------------------------------------------------
// Kernel 1: proj{2,3}[b,w,:] = input2[b,w,:] @ W{2,3}^T + b{2,3}
// One wave per 16(w) x 16(j) tile, both heads at once (A fragments shared).
// NT gemm: A(m,k)=attn[w0+m][k], B(k,n)=W[j0+n][k] -> both are float2 loads.
// ---------------------------------------------------------------------------
__global__ __launch_bounds__(256) void proj_kernel(
    const float* __restrict__ attn,
    const float* __restrict__ W2, const float* __restrict__ b2,
    const float* __restrict__ W3, const float* __restrict__ b3,
    float* __restrict__ proj2, float* __restrict__ proj3)
{
    const int lane = threadIdx.x & 31;
    const int wv   = threadIdx.x >> 5;
    const int tile = blockIdx.x * 8 + wv;      // 9728 tiles total, grid sized exactly
    const int jt = tile % 19;
    const int wt = (tile / 19) % 32;
    const int b  = tile / (19 * 32);

    const int m = lane & 15;                   // row-in-tile / col-in-tile (n == m)
    const int h = lane >> 4;
    const int w0 = wt * 16, j0 = jt * 16;

    const int j  = j0 + m;                     // output column (W row)
    const int jc = (j < Dc) ? j : (Dc - 1);    // clamped: in-bounds, unmasked

    const float* attnb = attn + (size_t)b * WNc * Dc;
    const float* aRow  = attnb + (w0 + m) * Dc + 2 * h;
    const float* w2Row = W2 + jc * Dc + 2 * h;
    const float* w3Row = W3 + jc * Dc + 2 * h;

    v8f c2 = {}; v8f c3 = {};
#pragma unroll 5
    for (int k0 = 0; k0 < Dc; k0 += 4) {       // 75 K-steps, pure load+wmma
        v2f a   = *(const v2f*)(aRow + k0);
        v2f bb2 = *(const v2f*)(w2Row + k0);
        v2f bb3 = *(const v2f*)(w3Row + k0);
        c2 = __builtin_amdgcn_wmma_f32_16x16x4_f32(false, a, false, bb2, (short)0, c2, false, false);
        c3 = __builtin_amdgcn_wmma_f32_16x16x4_f32(false, a, false, bb3, (short)0, c3, false, false);
    }

    const float bias2 = b2[jc];
    const float bias3 = b3[jc];
    float* p2 = proj2 + (size_t)b * WNc * Dc;
    float* p3 = proj3 + (size_t)b * WNc * Dc;
    if (j < Dc) {                              // guard kills the garbage columns
#pragma unroll
        for (int g = 0; g < 8; ++g) {          // C/D layout: vgpr g -> row g+8h, col n
            const int w = w0 + g + 8 * h;
            p2[w * Dc + j] = c2[g] + bias2;
            p3[w * Dc + j] = c3[g] + bias3;
        }
    }
}

// ---------------------------------------------------------------------------
// Kernel 2: per (b, 16-row t-tile, head):
//   scores[16,512] = mainTile @ proj^T   (WMMA, LDS-staged A)
//   probs = softmax(scores)              (wave32 shfl reductions, v_exp_f32)
//   out[16,300]   = probs @ attn         (WMMA)
// ---------------------------------------------------------------------------
__global__ __launch_bounds__(512) void attend_kernel(
    const float* __restrict__ main_,
    const float* __restrict__ attn,
    const float* __restrict__ proj0,
    float* __restrict__ out)
{
    __shared__ float sMain[16 * Dc];           // 19,200 B
    __shared__ float sP[16 * SST];             // 33,024 B  (scores -> probs in place)

    const int tt   = blockIdx.x;               // t-tile 0..127
    const int b    = blockIdx.y;               // 0..15
    const int head = blockIdx.z;               // 0..1
    const int tid  = threadIdx.x;
    const int lane = tid & 31;
    const int wv   = tid >> 5;                 // 16 waves
    const int m = lane & 15, h = lane >> 4;
    const int t0 = tt * 16;

    const float* mainb = main_ + ((size_t)b * Tc + t0) * Dc;
    const float* attnb = attn  + (size_t)b * WNc * Dc;
    const float* proj  = proj0 + (size_t)head * Bc * WNc * Dc + (size_t)b * WNc * Dc;
    float*       outb  = out   + (size_t)head * Bc * Tc * Dc + ((size_t)b * Tc + t0) * Dc;

    // ---- stage main tile (16 rows are contiguous: 4800 floats) ----
    for (int idx = tid; idx < 16 * Dc; idx += 512) sMain[idx] = mainb[idx];
    __syncthreads();

    // ---- phase 1: scores. 32 w-tiles over 16 waves ----
    for (int wt = wv; wt < 32; wt += 16) {
        const int w0 = wt * 16;
        const float* aBase = sMain + m * Dc + 2 * h;          // A(m,k)=main[m][k]
        const float* bBase = proj + (w0 + m) * Dc + 2 * h;    // B(k,n)=proj[w0+n][k], n==m
        v8f acc = {};
#pragma unroll 5
        for (int k0 = 0; k0 < Dc; k0 += 4) {                  // 75 K-steps
            v2f a  = *(const v2f*)(aBase + k0);
            v2f bb = *(const v2f*)(bBase + k0);
            acc = __builtin_amdgcn_wmma_f32_16x16x4_f32(false, a, false, bb, (short)0, acc, false, false);
        }
#pragma unroll
        for (int g = 0; g < 8; ++g)
            sP[(g + 8 * h) * SST + w0 + m] = acc[g];
    }
    __syncthreads();

    // ---- phase 2: softmax over 512, one row per wave ----
    {
        float v[16];
        float mx = -3.402823466e+38f;
#pragma unroll
        for (int i = 0; i < 16; ++i) { v[i] = sP[wv * SST + lane + 32 * i]; mx = fmaxf(mx, v[i]); }
#pragma unroll
        for (int off = 16; off > 0; off >>= 1) mx = fmaxf(mx, __shfl_xor(mx, off, 32));
        float s = 0.0f;
#pragma unroll
        for (int i = 0; i < 16; ++i) { v[i] = __expf(v[i] - mx); s += v[i]; }
#pragma unroll
        for (int off = 16; off > 0; off >>= 1) s += __shfl_xor(s, off, 32);
        const float inv = 1.0f / s;
#pragma unroll
        for (int i = 0; i < 16; ++i) sP[wv * SST + lane + 32 * i] = v[i] * inv;
    }
    __syncthreads();

    // ---- phase 3: out = probs @ attn. 19 n-tiles over 16 waves ----
    for (int nt = wv; nt < 19; nt += 16) {
        const int n0   = nt * 16;
        const int col  = n0 + m;
        const int colc = (col < Dc) ? col : (Dc - 1);         // clamped, unmasked
        const float* aBase = sP + m * SST + 2 * h;            // A(m,k)=probs[m][k]
        const float* bBase = attnb + colc;                    // B(k,n)=attn[k][n0+n]
        v8f acc = {};
#pragma unroll 4
        for (int k0 = 0; k0 < WNc; k0 += 4) {                 // 128 K-steps
            v2f a = *(const v2f*)(aBase + k0);
            const int kr = k0 + 2 * h;
            v2f bb;
            bb.x = bBase[(size_t)kr * Dc];
            bb.y = bBase[(size_t)(kr + 1) * Dc];
            acc = __builtin_amdgcn_wmma_f32_16x16x4_f32(false, a, false, bb, (short)0, acc, false, false);
        }
        if (col < Dc) {                                       // guard kills garbage columns
#pragma unroll
            for (int g = 0; g < 8; ++g)
                outb[(g + 8 * h) * Dc + col] = acc[g];
        }
    }
}

// ---------------------------------------------------------------------------
extern "C" void kernel_launch(void* const* d_in, const int* in_sizes, int n_in,
                              void* d_out, int out_size, void* d_ws, size_t ws_size,
                              hipStream_t stream)
{
    const float* input1 = (const float*)d_in[0];   // [16,2048,300]
    const float* input2 = (const float*)d_in[1];   // [16,512,300]
    const float* W2     = (const float*)d_in[2];   // [300,300]
    const float* b2     = (const float*)d_in[3];   // [300]
    const float* W3     = (const float*)d_in[4];   // [300,300]
    const float* b3     = (const float*)d_in[5];   // [300]
    // d_in[6] = mode (setup uses mode==1: compute both heads)

    float* outp = (float*)d_out;                   // [2][16][2048][300] concatenated
    float* proj = (float*)d_ws;                    // [2][16][512][300] scratch (19.7 MB)
    const size_t projElems = (size_t)Bc * WNc * Dc;

    // 16 batches * 32 w-tiles * 19 j-tiles = 9728 waves; 8 waves/block -> 1216 blocks exact
    proj_kernel<<<dim3(1216), dim3(256), 0, stream>>>(input2, W2, b2, W3, b3,
                                                      proj, proj + projElems);
    attend_kernel<<<dim3(128, 16, 2), dim3(512), 0, stream>>>(input1, input2, proj, outp);

    (void)in_sizes; (void)n_in; (void)out_size; (void)ws_size;
}